// GATLayer_82867099009054
// MI455X (gfx1250) — compile-verified
//
#include <hip/hip_runtime.h>
#include <math.h>

#define N_NODES 100000
#define N_EDGES 1600000
#define IN_DIM  128
#define OUT_DIM 64
#define ALPHA   0.2f
#define BN_EPS  1e-5f

// workspace layout (float offsets)
#define OFF_MSG   ((size_t)0)         // N*64 message
#define OFF_ASRC  ((size_t)6400000)   // N attn_src
#define OFF_ADST  ((size_t)6500000)   // N attn_dst
#define OFF_EXPS  ((size_t)6600000)   // E exp scores
#define OFF_ESUM  ((size_t)8200000)   // N segment sums
#define OFF_BN    ((size_t)8300000)   // [0:64) sum [64:128) sumsq [128:192) scale [192:256) shift
#define OFF_GMAX  ((size_t)8300256)   // 1 x unsigned (ordered-encoded float max)

// Transposed W tile in LDS: sWt[n * LDS_STRIDE + k].
// Stride 132 dwords: 132 % 64 == 4, so a ds_load_b64 by 32 lanes covers all 64
// banks exactly once (half-wave 0 -> banks {4c+k, 4c+k+1}, half-wave 1 -> {4c+k+2, 4c+k+3}).
#define LDS_STRIDE 132

typedef __attribute__((ext_vector_type(2))) float v2f;
typedef __attribute__((ext_vector_type(8))) float v8f;

// order-preserving float<->uint encoding so atomicMax(uint) == float max (handles negatives)
__device__ __forceinline__ unsigned enc_f32(float f) {
  unsigned u = __float_as_uint(f);
  return (u & 0x80000000u) ? ~u : (u | 0x80000000u);
}
__device__ __forceinline__ float dec_f32(unsigned e) {
  return (e & 0x80000000u) ? __uint_as_float(e & 0x7fffffffu) : __uint_as_float(~e);
}

__global__ void k_init(float* __restrict__ out, float* __restrict__ ws) {
  size_t i = (size_t)blockIdx.x * 256 + threadIdx.x;
  if (i < (size_t)N_NODES * OUT_DIM) out[i] = 0.0f;
  if (i < (size_t)N_NODES)           ws[OFF_ESUM + i] = 0.0f;
  if (i < 256)                       ws[OFF_BN + i] = 0.0f;
  if (i == 0) ((unsigned*)(ws + OFF_GMAX))[0] = 0u;  // encodes "minimum"
}

// message = feat @ W via V_WMMA_F32_16X16X4_F32; one wave -> 16 rows x 64 cols.
// W staged TRANSPOSED in LDS so each B fragment is a single aligned ds_load_b64.
__global__ __launch_bounds__(256) void k_gemm(const float* __restrict__ feat,
                                              const float* __restrict__ W,
                                              float* __restrict__ msg) {
  __shared__ float sWt[OUT_DIM * LDS_STRIDE];  // ~33 KB of 320 KB WGP LDS
  int tid = threadIdx.x;
  for (int i = tid; i < IN_DIM * OUT_DIM; i += 256) {
    int k = i >> 6;   // W row   (0..127)
    int n = i & 63;   // W col   (0..63)
    sWt[n * LDS_STRIDE + k] = W[i];
  }
  __syncthreads();

  int wave = tid >> 5, lane = tid & 31;
  int m0 = blockIdx.x * 128 + wave * 16;
  int mrow = m0 + (lane & 15);
  if (mrow >= N_NODES) mrow = N_NODES - 1;          // clamp loads; stores masked below
  const float* fr = feat + (size_t)mrow * IN_DIM;
  int koff = (lane < 16) ? 0 : 2;                   // f32 A/B fragment K split per ISA 7.12.2
  int col  = lane & 15;

  v8f acc[4] = {};
  for (int k0 = 0; k0 < IN_DIM; k0 += 4) {
    v2f a = *(const v2f*)(fr + k0 + koff);          // A: feat[m][k], feat[m][k+1]
#pragma unroll
    for (int t = 0; t < 4; ++t) {
      // B: W[k][n], W[k+1][n] contiguous in transposed LDS -> one ds_load_b64
      v2f b = *(const v2f*)&sWt[(col + t * 16) * LDS_STRIDE + k0 + koff];
      acc[t] = __builtin_amdgcn_wmma_f32_16x16x4_f32(
          false, a, false, b, (short)0, acc[t], false, false);
    }
  }

  int mbase = m0 + ((lane >> 4) << 3);  // lanes 16-31 hold M = v+8
#pragma unroll
  for (int t = 0; t < 4; ++t) {
#pragma unroll
    for (int v = 0; v < 8; ++v) {
      int m = mbase + v;
      if (m < N_NODES) msg[(size_t)m * OUT_DIM + t * 16 + col] = acc[t][v];
    }
  }
}

// attn_src/dst = message . a[:64] / a[64:]; 16 lanes per node, float4 loads
__global__ void k_attn(const float* __restrict__ msg, const float* __restrict__ avec,
                       float* __restrict__ asrc, float* __restrict__ adst) {
  int l = threadIdx.x & 15;
  int node = blockIdx.x * 16 + (threadIdx.x >> 4);
  if (node >= N_NODES) return;
  float4 m  = ((const float4*)(msg + (size_t)node * OUT_DIM))[l];
  float4 a0 = ((const float4*)avec)[l];
  float4 a1 = ((const float4*)(avec + OUT_DIM))[l];
  float s = m.x * a0.x + m.y * a0.y + m.z * a0.z + m.w * a0.w;
  float d = m.x * a1.x + m.y * a1.y + m.z * a1.z + m.w * a1.w;
#pragma unroll
  for (int off = 8; off; off >>= 1) {
    s += __shfl_xor(s, off, 32);
    d += __shfl_xor(d, off, 32);
  }
  if (l == 0) { asrc[node] = s; adst[node] = d; }
}

__global__ void k_edge_max(const int* __restrict__ src, const int* __restrict__ dst,
                           const float* __restrict__ asrc, const float* __restrict__ adst,
                           unsigned* __restrict__ gmax) {
  float m = -3.402823466e38f;
  int stride = gridDim.x * blockDim.x;
  for (int e = blockIdx.x * blockDim.x + threadIdx.x; e < N_EDGES; e += stride) {
    if (e + stride < N_EDGES) {
      __builtin_prefetch(&src[e + stride], 0, 0);   // global_prefetch_b8
      __builtin_prefetch(&dst[e + stride], 0, 0);
    }
    float x = asrc[src[e]] + adst[dst[e]];
    x = (x > 0.0f) ? x : ALPHA * x;   // leaky relu
    m = fmaxf(m, x);
  }
#pragma unroll
  for (int off = 16; off; off >>= 1) m = fmaxf(m, __shfl_xor(m, off, 32));
  if ((threadIdx.x & 31) == 0) atomicMax(gmax, enc_f32(m));
}

__global__ void k_edge_exp(const int* __restrict__ src, const int* __restrict__ dst,
                           const float* __restrict__ asrc, const float* __restrict__ adst,
                           const unsigned* __restrict__ gmax,
                           float* __restrict__ exps, float* __restrict__ esum) {
  int e = blockIdx.x * 256 + threadIdx.x;
  if (e >= N_EDGES) return;
  float gm = dec_f32(*gmax);
  float x = asrc[src[e]] + adst[dst[e]];
  x = (x > 0.0f) ? x : ALPHA * x;
  float s = __expf(x - gm);
  exps[e] = s;
  atomicAdd(&esum[dst[e]], s);
}

// out[dst] += message[src] * (exp/expsum); 16 lanes per edge, float4 gathers, L2-resident atomics
__global__ void k_edge_scatter(const int* __restrict__ src, const int* __restrict__ dst,
                               const float* __restrict__ exps, const float* __restrict__ esum,
                               const float* __restrict__ msg, float* __restrict__ out) {
  int e = blockIdx.x * 16 + (threadIdx.x >> 4);
  int l = threadIdx.x & 15;
  if (e >= N_EDGES) return;
  int s = src[e], d = dst[e];
  float coeff = exps[e] / (esum[d] + 1e-10f);
  float4 m = ((const float4*)(msg + (size_t)s * OUT_DIM))[l];
  float* o = out + (size_t)d * OUT_DIM + (size_t)l * 4;
  atomicAdd(o + 0, m.x * coeff);
  atomicAdd(o + 1, m.y * coeff);
  atomicAdd(o + 2, m.z * coeff);
  atomicAdd(o + 3, m.w * coeff);
}

__global__ __launch_bounds__(256) void k_bn_reduce(const float* __restrict__ out,
                                                   float* __restrict__ bn) {
  __shared__ float ssum[64], ssq[64];
  int tid = threadIdx.x;
  if (tid < 64) { ssum[tid] = 0.0f; ssq[tid] = 0.0f; }
  __syncthreads();
  int c = tid & 63;
  float sum = 0.0f, sq = 0.0f;
#pragma unroll 4
  for (int i = 0; i < 32; ++i) {
    int r = blockIdx.x * 128 + (tid >> 6) + i * 4;
    if (r < N_NODES) {
      float v = out[(size_t)r * OUT_DIM + c];
      sum += v; sq += v * v;
    }
  }
  atomicAdd(&ssum[c], sum);
  atomicAdd(&ssq[c], sq);
  __syncthreads();
  if (tid < 64) { atomicAdd(&bn[tid], ssum[tid]); atomicAdd(&bn[64 + tid], ssq[tid]); }
}

__global__ void k_bn_final(float* __restrict__ bn, const float* __restrict__ gamma,
                           const float* __restrict__ beta) {
  int c = threadIdx.x;
  if (c >= 64) return;
  const float invN = 1.0f / (float)N_NODES;
  float mu  = bn[c] * invN;
  float var = bn[64 + c] * invN - mu * mu;
  var = fmaxf(var, 0.0f);
  float inv = rsqrtf(var + BN_EPS);
  float scale = gamma[c] * inv;
  bn[128 + c] = scale;
  bn[192 + c] = beta[c] - mu * scale;
}

__global__ void k_bn_apply(float* __restrict__ out, const float* __restrict__ bn) {
  size_t i = (size_t)blockIdx.x * 256 + threadIdx.x;
  if (i >= (size_t)N_NODES * OUT_DIM) return;
  int c = (int)(i & 63);
  out[i] = out[i] * bn[128 + c] + bn[192 + c];
}

extern "C" void kernel_launch(void* const* d_in, const int* in_sizes, int n_in,
                              void* d_out, int out_size, void* d_ws, size_t ws_size,
                              hipStream_t stream) {
  const float* feat  = (const float*)d_in[0];
  const int*   edges = (const int*)d_in[1];
  const float* W     = (const float*)d_in[2];
  const float* avec  = (const float*)d_in[3];
  const float* gamma = (const float*)d_in[4];
  const float* beta  = (const float*)d_in[5];
  float* out = (float*)d_out;
  float* ws  = (float*)d_ws;
  const int* src = edges;             // edges[0,:]
  const int* dst = edges + N_EDGES;   // edges[1,:]

  k_init<<<25000, 256, 0, stream>>>(out, ws);
  k_gemm<<<(N_NODES + 127) / 128, 256, 0, stream>>>(feat, W, ws + OFF_MSG);
  k_attn<<<(N_NODES + 15) / 16, 256, 0, stream>>>(ws + OFF_MSG, avec,
                                                  ws + OFF_ASRC, ws + OFF_ADST);
  k_edge_max<<<1024, 256, 0, stream>>>(src, dst, ws + OFF_ASRC, ws + OFF_ADST,
                                       (unsigned*)(ws + OFF_GMAX));
  k_edge_exp<<<(N_EDGES + 255) / 256, 256, 0, stream>>>(src, dst, ws + OFF_ASRC,
                                                        ws + OFF_ADST,
                                                        (const unsigned*)(ws + OFF_GMAX),
                                                        ws + OFF_EXPS, ws + OFF_ESUM);
  k_edge_scatter<<<(N_EDGES + 15) / 16, 256, 0, stream>>>(src, dst, ws + OFF_EXPS,
                                                          ws + OFF_ESUM, ws + OFF_MSG, out);
  k_bn_reduce<<<(N_NODES + 127) / 128, 256, 0, stream>>>(out, ws + OFF_BN);
  k_bn_final<<<1, 64, 0, stream>>>(ws + OFF_BN, gamma, beta);
  k_bn_apply<<<25000, 256, 0, stream>>>(out, ws + OFF_BN);
}